// AttentionAggregationV2_63806034149756
// MI455X (gfx1250) — compile-verified
//
#include <hip/hip_runtime.h>
#include <cstdint>

// AttentionAggregationV2: edge softmax + weighted scatter-sum.
// E=1.6M edges, H=8 heads, head_dim=6, D=48, N=50K nodes.
// Memory-bound: ~370MB streamed -> ~17us floor at 23.3 TB/s HBM.
// CDNA5 paths: global_load_async_to_lds_b128 (NT-hinted, + s_wait_asynccnt)
// for coalesced value staging into LDS; hardware global_atomic_add_f32 for
// segment sums / scatter (executes in the 192MB L2, which the NT hints keep
// free for the 11MB hot out/seg buffers).

#define NUM_HEADS 8
#define DVAL      48
#define HEAD_DIM  6
#define EPB       256            // edges per block in scatter kernel
#define ROWF      52             // padded LDS row stride in floats (208B, 16B-mult)

typedef float v4f __attribute__((ext_vector_type(4)));

// ----------------------------------------------------------- scatter --------
// Block stages value[e0 : e0+256, 0:48] (48KB) into LDS via async b128 copies
// (contiguous global stream, padded LDS rows -> <=2-way bank conflicts), then
// each thread computes its edge's 8 alphas and scatter-adds 48 floats.
__global__ void __launch_bounds__(EPB)
k_scatter(const float* __restrict__ value,
          const float* __restrict__ wt,
          const float* __restrict__ cut,
          const int*   __restrict__ dst,
          const float* __restrict__ seg,
          float* __restrict__ out, int E) {
  __shared__ __align__(16) float smem[EPB * ROWF];

  const int t  = threadIdx.x;
  const int e0 = blockIdx.x * EPB;
  const int nEdge = min(EPB, E - e0);

  // ---- async stage: 12 x 16B chunks per row, chunk ids contiguous in memory
  {
    const unsigned long long gbase =
        (unsigned long long)(const void*)(value + (size_t)e0 * DVAL);
    const unsigned lds_base = (unsigned)(uintptr_t)(&smem[0]);
    const int maxChunk = nEdge * 12;
#pragma unroll
    for (int it = 0; it < 12; ++it) {
      int i = it * EPB + t;                       // global 16B-chunk index
      if (i < maxChunk) {
        int row = i / 12;
        int c   = i - row * 12;
        unsigned lds_off = lds_base + (unsigned)(row * (ROWF * 4) + c * 16);
        unsigned goff    = (unsigned)i * 16u;
        asm volatile("global_load_async_to_lds_b128 %0, %1, %2 th:TH_LOAD_NT"
                     :: "v"(lds_off), "v"(goff), "s"(gbase) : "memory");
      }
    }
  }
  asm volatile("s_wait_asynccnt 0" ::: "memory");
  __syncthreads();

  const int e = e0 + t;
  if (e >= E) return;

  const float c = __builtin_nontemporal_load(cut + e);
  const int   d = __builtin_nontemporal_load(dst + e);

  // logits: two coalesced b128 NT loads per thread (contiguous 32B, streamed)
  const v4f* wrow = (const v4f*)(wt + (size_t)e * NUM_HEADS);
  v4f w0 = __builtin_nontemporal_load(wrow + 0);
  v4f w1 = __builtin_nontemporal_load(wrow + 1);
  // segment sums for this node: hot L2-resident gather, keep regular-temporal
  const v4f* srow = (const v4f*)(seg + (size_t)d * NUM_HEADS);
  v4f s0 = srow[0];
  v4f s1 = srow[1];

  float al[NUM_HEADS];
  al[0] = __expf(c * w0.x) / s0.x;
  al[1] = __expf(c * w0.y) / s0.y;
  al[2] = __expf(c * w0.z) / s0.z;
  al[3] = __expf(c * w0.w) / s0.w;
  al[4] = __expf(c * w1.x) / s1.x;
  al[5] = __expf(c * w1.y) / s1.y;
  al[6] = __expf(c * w1.z) / s1.z;
  al[7] = __expf(c * w1.w) / s1.w;

  const float* v = &smem[t * ROWF];
  float* orow = out + (size_t)d * DVAL;
#pragma unroll
  for (int h = 0; h < NUM_HEADS; ++h) {
    float a = al[h];
#pragma unroll
    for (int k = 0; k < HEAD_DIM; ++k) {
      unsafeAtomicAdd(&orow[h * HEAD_DIM + k], a * v[h * HEAD_DIM + k]);
    }
  }
}

// ------------------------------------------------------- segment exp-sum ----
// One thread per (edge, head). Lanes 0..7 share one edge -> cutoff/dst loads
// broadcast; edge_weights loads are perfectly coalesced. All inputs streamed
// once -> non-temporal.
__global__ void k_segsum(const float* __restrict__ wt,
                         const float* __restrict__ cut,
                         const int*   __restrict__ dst,
                         float* __restrict__ seg, int EH) {
  int tid = blockIdx.x * blockDim.x + threadIdx.x;
  if (tid >= EH) return;
  int e = tid >> 3;
  int h = tid & 7;
  float wv = __builtin_nontemporal_load(wt + tid);
  float cv = __builtin_nontemporal_load(cut + e);
  int   d  = __builtin_nontemporal_load(dst + e);
  float ew = __expf(cv * wv);                // softmax is shift-invariant; |w|<~6
  unsafeAtomicAdd(&seg[(size_t)d * NUM_HEADS + h], ew);  // global_atomic_add_f32
}

// ---------------------------------------------------------------- init ------
// Exact-size b128 zero stores; both sizes are multiples of 4 floats.
__global__ void k_zero(v4f* __restrict__ out4, v4f* __restrict__ seg4,
                       int n4_out, int n4_seg) {
  int i = blockIdx.x * blockDim.x + threadIdx.x;
  v4f z = (v4f)(0.0f);
  if (i < n4_out) out4[i] = z;
  if (i < n4_seg) seg4[i] = z;
}

// ---------------------------------------------------------------- launch ----
extern "C" void kernel_launch(void* const* d_in, const int* in_sizes, int n_in,
                              void* d_out, int out_size, void* d_ws, size_t ws_size,
                              hipStream_t stream) {
  const float* value = (const float*)d_in[0];   // [E,48]
  const float* wt    = (const float*)d_in[1];   // [E,8]
  const float* cut   = (const float*)d_in[2];   // [E]
  const int*   eidx  = (const int*)d_in[3];     // [2,E] (harness: integer -> int32)

  const int E = in_sizes[2];
  const int N = out_size / DVAL;
  const int* dst = eidx + E;                    // edge_index[1]

  float* out = (float*)d_out;                   // [N,48]
  float* seg = (float*)d_ws;                    // [N,8] scratch (1.6MB)

  const int EH = E * NUM_HEADS;
  const int n4_out = (N * DVAL) / 4;
  const int n4_seg = (N * NUM_HEADS) / 4;
  const int n4_max = n4_out > n4_seg ? n4_out : n4_seg;

  k_zero<<<(n4_max + 255) / 256, 256, 0, stream>>>(
      (v4f*)out, (v4f*)seg, n4_out, n4_seg);
  k_segsum<<<(EH + 255) / 256, 256, 0, stream>>>(wt, cut, dst, seg, EH);
  k_scatter<<<(E + EPB - 1) / EPB, EPB, 0, stream>>>(value, wt, cut, dst, seg, out, E);
}